// Warping_layer_23802708755023
// MI455X (gfx1250) — compile-verified
//
#include <hip/hip_runtime.h>

// Problem constants from the reference: x[B=8, C=32, H=512, W=512], flow[B,2,H,W]
constexpr int Bn = 8;
constexpr int Cn = 32;
constexpr int Hn = 512;
constexpr int Wn = 512;
constexpr int HW = Hn * Wn;

// Bilinear backward-warp with zeros padding, align_corners=True,
// grid = base_grid - flow/((size-1)/2)  (exactly as the reference).
//
// One thread per output pixel (b,h,w); loops over all C=32 channels so the
// flow load + coordinate/weight math is amortized. Lanes span consecutive w
// -> corner gathers are near-contiguous per wave (L2-friendly). Output is
// written with non-temporal stores to avoid polluting L2, which we want to
// keep for the heavily re-used x rows.
__global__ __launch_bounds__(256) void warp_bilinear_kernel(
    const float* __restrict__ x,
    const float* __restrict__ flow,
    float* __restrict__ out)
{
    const int pix = blockIdx.x * 256 + threadIdx.x;   // 0 .. B*H*W-1
    if (pix >= Bn * HW) return;

    const int w = pix & (Wn - 1);
    const int t = pix >> 9;            // / Wn
    const int h = t & (Hn - 1);
    const int b = t >> 9;              // / Hn

    // flow[b,0,h,w] and flow[b,1,h,w]
    const int fbase = (b * 2) * HW + h * Wn + w;
    const float flx = flow[fbase];
    const float fly = flow[fbase + HW];

    // Mirror the reference float sequence:
    //   fx = flow / ((W-1)/2);  gx = linspace(-1,1,W)[w] - fx
    //   ix = (gx + 1) * 0.5 * (W-1)
    const float sx = 2.0f / (float)(Wn - 1);
    const float sy = 2.0f / (float)(Hn - 1);
    const float gx = (-1.0f + (float)w * sx) - flx * sx;
    const float gy = (-1.0f + (float)h * sy) - fly * sy;
    const float ix = (gx + 1.0f) * (0.5f * (float)(Wn - 1));
    const float iy = (gy + 1.0f) * (0.5f * (float)(Hn - 1));

    const float x0f = floorf(ix);
    const float y0f = floorf(iy);
    const float wx1 = ix - x0f;
    const float wy1 = iy - y0f;
    const float wx0 = 1.0f - wx1;
    const float wy0 = 1.0f - wy1;
    const float x1f = x0f + 1.0f;
    const float y1f = y0f + 1.0f;

    // zeros padding: out-of-bounds corner contributes 0 (validity folded into
    // the weight, loads always use clamped in-bounds addresses).
    const float vx0 = (x0f >= 0.0f && x0f < (float)Wn) ? 1.0f : 0.0f;
    const float vx1 = (x1f >= 0.0f && x1f < (float)Wn) ? 1.0f : 0.0f;
    const float vy0 = (y0f >= 0.0f && y0f < (float)Hn) ? 1.0f : 0.0f;
    const float vy1 = (y1f >= 0.0f && y1f < (float)Hn) ? 1.0f : 0.0f;

    const float wxa = wx0 * vx0;
    const float wxb = wx1 * vx1;
    const float wya = wy0 * vy0;
    const float wyb = wy1 * vy1;
    const float w00 = wya * wxa;
    const float w01 = wya * wxb;
    const float w10 = wyb * wxa;
    const float w11 = wyb * wxb;

    const int X0 = (int)fminf(fmaxf(x0f, 0.0f), (float)(Wn - 1));
    const int X1 = (int)fminf(fmaxf(x1f, 0.0f), (float)(Wn - 1));
    const int Y0 = (int)fminf(fmaxf(y0f, 0.0f), (float)(Hn - 1));
    const int Y1 = (int)fminf(fmaxf(y1f, 0.0f), (float)(Hn - 1));

    const int o00 = Y0 * Wn + X0;
    const int o01 = Y0 * Wn + X1;
    const int o10 = Y1 * Wn + X0;
    const int o11 = Y1 * Wn + X1;

    const float* __restrict__ xb = x + (size_t)b * Cn * HW;
    float* __restrict__ ob = out + (size_t)b * Cn * HW + (size_t)(h * Wn + w);

    // 32 channels; unroll so many independent global_load_b32 are in flight.
    #pragma unroll 8
    for (int c = 0; c < Cn; ++c) {
        const float* __restrict__ xc = xb + (size_t)c * HW;
        float v = xc[o00] * w00;
        v = fmaf(xc[o01], w01, v);
        v = fmaf(xc[o10], w10, v);
        v = fmaf(xc[o11], w11, v);
        // streaming output: non-temporal store (TH_NT) to preserve L2 for x
        __builtin_nontemporal_store(v, ob + (size_t)c * HW);
    }
}

extern "C" void kernel_launch(void* const* d_in, const int* in_sizes, int n_in,
                              void* d_out, int out_size, void* d_ws, size_t ws_size,
                              hipStream_t stream) {
    (void)in_sizes; (void)n_in; (void)out_size; (void)d_ws; (void)ws_size;
    const float* x    = (const float*)d_in[0];   // [B,C,H,W] fp32
    const float* flow = (const float*)d_in[1];   // [B,2,H,W] fp32
    float* out        = (float*)d_out;           // [B,C,H,W] fp32

    const int total  = Bn * HW;                  // one thread per (b,h,w) pixel
    const int block  = 256;                      // 8 wave32 per block
    const int blocks = (total + block - 1) / block;
    warp_bilinear_kernel<<<blocks, block, 0, stream>>>(x, flow, out);
}